// LSGCL_54030688583861
// MI455X (gfx1250) — compile-verified
//
#include <hip/hip_runtime.h>
#include <math.h>

// ---------------------------------------------------------------------------
// LSGCL forward on MI455X (gfx1250, wave32).
//
//  deg/dinv  : atomics over edges (+1 self loop), rsqrt
//  3 hops    : xn[i] = dinv[i]^2 * x[i] + sum_{e:(j->i)} dinv[j]dinv[i] * x[j]
//              (one wave32 per edge, float4 gather, global_atomic_add_f32 scatter,
//               prefetch of next edge's source row)
//  4 GEMMs   : t = x @ W[k]^T + b[k], fused row L2-norm, via
//              v_wmma_f32_16x16x4_f32 (fp32 WMMA, 16x16 tiles, K-step 4).
//              W staged row-major in LDS with 68-float row pitch so each B
//              fragment is ONE aligned, bank-conflict-free ds_load_b64.
// ---------------------------------------------------------------------------

typedef __attribute__((ext_vector_type(2))) float v2f;
typedef __attribute__((ext_vector_type(8))) float v8f;

#define FEAT 128
#define ODIM 512
#define WPITCH 68   // 64-col chunk + 4 pad floats: aligned float4 stores,
                    // conflict-free b64 reads (stride 4 banks/lane)

// ---------------- degree / dinv ----------------

__global__ void lsgcl_deg_init(float* __restrict__ deg, int n) {
    int i = blockIdx.x * blockDim.x + threadIdx.x;
    if (i < n) deg[i] = 1.0f;               // self loop
}

__global__ void lsgcl_deg_count(float* __restrict__ deg,
                                const int* __restrict__ ecol, int E) {
    int e = blockIdx.x * blockDim.x + threadIdx.x;
    if (e < E) atomicAdd(deg + ecol[e], 1.0f);
}

__global__ void lsgcl_dinv(float* __restrict__ deg, int n) {
    int i = blockIdx.x * blockDim.x + threadIdx.x;
    if (i < n) {
        float d = deg[i];
        deg[i] = (d > 0.0f) ? rsqrtf(d) : 0.0f;   // in-place -> dinv
    }
}

// ---------------- self-loop init: xn = dinv^2 * x ----------------

__global__ void lsgcl_selfloop(const float* __restrict__ x,
                               float* __restrict__ xn,
                               const float* __restrict__ dinv, int n) {
    size_t idx = (size_t)blockIdx.x * blockDim.x + threadIdx.x;  // over n*32 float4s
    size_t total = (size_t)n * (FEAT / 4);
    if (idx >= total) return;
    int node = (int)(idx >> 5);
    float w = dinv[node];
    w = w * w;
    float4 v = ((const float4*)x)[idx];
    float4 r;
    r.x = v.x * w; r.y = v.y * w; r.z = v.z * w; r.w = v.w * w;
    ((float4*)xn)[idx] = r;
}

// ---------------- edge scatter: one wave32 per edge ----------------

__global__ __launch_bounds__(256)
void lsgcl_scatter(const float* __restrict__ x, float* __restrict__ xn,
                   const int* __restrict__ erow, const int* __restrict__ ecol,
                   const float* __restrict__ dinv, int E) {
    int gid = blockIdx.x * blockDim.x + threadIdx.x;
    int waveId = gid >> 5;
    int lane = threadIdx.x & 31;
    int numWaves = (gridDim.x * blockDim.x) >> 5;

    for (int e = waveId; e < E; e += numWaves) {
        int j = erow[e];
        int i = ecol[e];
        // Prefetch the next edge's source row into cache (global_prefetch_b8).
        int en = e + numWaves;
        if (en < E) {
            int jn = erow[en];
            __builtin_prefetch(x + (size_t)jn * FEAT + lane * 4, 0, 1);
        }
        float w = dinv[j] * dinv[i];
        float4 v = ((const float4*)(x + (size_t)j * FEAT))[lane];
        float* dst = xn + (size_t)i * FEAT + lane * 4;
        atomicAdd(dst + 0, v.x * w);
        atomicAdd(dst + 1, v.y * w);
        atomicAdd(dst + 2, v.z * w);
        atomicAdd(dst + 3, v.w * w);
    }
}

// ---------------- WMMA GEMM + bias + fused row L2-norm ----------------
// Block: 256 threads = 8 waves, each wave computes a 16(M) x 128(N) strip.
// Block covers 128 rows. W[k] (128x128) staged ROW-MAJOR (padded pitch 68)
// in LDS, two 64-wide K chunks: Wsh[h * WPITCH + (f - kc)].
// B fragment (W[n*128+kb], W[n*128+kb+1]) is then one aligned ds_load_b64.

__global__ __launch_bounds__(256)
void lsgcl_gemm_norm(const float* __restrict__ X,
                     const float* __restrict__ Wk,     // [128 x 128], row h, col f
                     const float* __restrict__ bk,     // [128]
                     const int* __restrict__ normFlag, // device scalar
                     float* __restrict__ out, int n, int slice) {
    __shared__ float Wsh[128 * WPITCH];   // 34 KB chunk

    const int tid = threadIdx.x;
    const int lane = tid & 31;
    const int wave = tid >> 5;
    const int hi = lane >> 4;        // 0: lanes 0-15, 1: lanes 16-31
    const int l16 = lane & 15;
    const int rowBase = blockIdx.x * 128 + wave * 16;

    // Clamp A-row for the tail block: WMMA requires EXEC all ones, so we load
    // a duplicate row instead of predicating; stores are guarded below.
    int arow = rowBase + l16;
    if (arow > n - 1) arow = n - 1;
    const float* aptr = X + (size_t)arow * FEAT + (hi ? 2 : 0);

    // Seed accumulators with the bias (C input of D = A*B + C).
    v8f acc[8];
#pragma unroll
    for (int t = 0; t < 8; ++t) {
        float bias = bk[t * 16 + l16];
        v8f c;
#pragma unroll
        for (int q = 0; q < 8; ++q) c[q] = bias;
        acc[t] = c;
    }

    for (int kc = 0; kc < FEAT; kc += 64) {
        __syncthreads();
        // Stage Wk[:, kc:kc+64] row-major into padded LDS.
        for (int i = tid; i < (128 * 64) / 4; i += 256) {
            int hrow = i >> 4;             // 16 float4 per 64-col row
            int f = (i & 15) * 4;
            float4 w4 = *(const float4*)(Wk + (size_t)hrow * FEAT + kc + f);
            *(float4*)(&Wsh[hrow * WPITCH + f]) = w4;   // aligned: 68h+f % 4 == 0
        }
        __syncthreads();

        for (int k0 = kc; k0 < kc + 64; k0 += 4) {
            // A fragment (16x4 fp32): lanes 0-15 hold (K0,K0+1), 16-31 (K0+2,K0+3)
            v2f a = *(const v2f*)(aptr + k0);
            const int kb = (k0 - kc) + (hi ? 2 : 0);
#pragma unroll
            for (int t = 0; t < 8; ++t) {
                const int ncol = t * 16 + l16;
                // B[k][n],B[k+1][n] = W[n*128+kb],W[n*128+kb+1]: contiguous.
                v2f bfrag = *(const v2f*)(&Wsh[ncol * WPITCH + kb]);
                acc[t] = __builtin_amdgcn_wmma_f32_16x16x4_f32(
                    false, a, false, bfrag, (short)0, acc[t], false, false);
            }
        }
    }

    // Fused row L2 norm. C/D layout: lane holds column N=l16; VGPR v is row
    // M=v (lanes 0-15) or M=v+8 (lanes 16-31) -> reduce within 16-lane half.
    const int doNorm = normFlag[0];
    float scale[8];
#pragma unroll
    for (int v = 0; v < 8; ++v) {
        float s = 0.0f;
#pragma unroll
        for (int t = 0; t < 8; ++t) {
            float e = acc[t][v];
            s += e * e;
        }
        s += __shfl_xor(s, 1, 32);
        s += __shfl_xor(s, 2, 32);
        s += __shfl_xor(s, 4, 32);
        s += __shfl_xor(s, 8, 32);
        float nrm = sqrtf(s);
        scale[v] = (doNorm == 1) ? (1.0f / fmaxf(nrm, 1e-12f)) : 1.0f;
    }

    const int colBase = slice * FEAT;
#pragma unroll
    for (int v = 0; v < 8; ++v) {
        int r = rowBase + v + (hi ? 8 : 0);
        if (r < n) {
            float* op = out + (size_t)r * ODIM + colBase + l16;
#pragma unroll
            for (int t = 0; t < 8; ++t) op[t * 16] = acc[t][v] * scale[v];
        }
    }
}

// ---------------------------------------------------------------------------

extern "C" void kernel_launch(void* const* d_in, const int* in_sizes, int n_in,
                              void* d_out, int out_size, void* d_ws, size_t ws_size,
                              hipStream_t stream) {
    const float* h  = (const float*)d_in[0];       // [N,128]
    const int* ei   = (const int*)d_in[1];         // [2,E]
    const int* nrm  = (const int*)d_in[2];         // scalar Norm
    const float* W  = (const float*)d_in[3];       // [4,128,128]
    const float* bb = (const float*)d_in[4];       // [4,128]
    float* out = (float*)d_out;

    const int N = in_sizes[0] / FEAT;
    const int E = in_sizes[1] / 2;
    const int* erow = ei;        // source j
    const int* ecol = ei + E;    // target i

    // Workspace: dinv (N) + two ping-pong feature buffers (N*128 each).
    float* dinv = (float*)d_ws;
    size_t npad = ((size_t)N + 255) & ~(size_t)255;
    float* bufA = dinv + npad;
    float* bufB = bufA + (size_t)N * FEAT;

    // Degrees + dinv
    lsgcl_deg_init<<<(N + 255) / 256, 256, 0, stream>>>(dinv, N);
    lsgcl_deg_count<<<(E + 255) / 256, 256, 0, stream>>>(dinv, ecol, E);
    lsgcl_dinv<<<(N + 255) / 256, 256, 0, stream>>>(dinv, N);

    const int gemmGrid = (N + 127) / 128;

    // k = 0: project h directly
    lsgcl_gemm_norm<<<gemmGrid, 256, 0, stream>>>(h, W, bb, nrm, out, N, 0);

    // Hops 1..3: selfloop init, edge scatter, project
    const float* cur = h;
    float* nxt = bufA;
    const size_t slTotal = ((size_t)N * (FEAT / 4) + 255) / 256;
    for (int k = 1; k <= 3; ++k) {
        lsgcl_selfloop<<<(int)slTotal, 256, 0, stream>>>(cur, nxt, dinv, N);
        lsgcl_scatter<<<4096, 256, 0, stream>>>(cur, nxt, erow, ecol, dinv, E);
        lsgcl_gemm_norm<<<gemmGrid, 256, 0, stream>>>(
            nxt, W + (size_t)k * FEAT * FEAT, bb + (size_t)k * FEAT, nrm, out, N, k);
        cur = nxt;
        nxt = (nxt == bufA) ? bufB : bufA;
    }
}